// Sinkhorn_3169685864479
// MI455X (gfx1250) — compile-verified
//
#include <hip/hip_runtime.h>
#include <hip/hip_bf16.h>

#define Bn 1024
#define Vn 4096

typedef __attribute__((ext_vector_type(2))) float v2f;
typedef __attribute__((ext_vector_type(8))) float v8f;

// ---------------------------------------------------------------------------
// 1) Row softmax of y/T (T=2).  blocks 0..1023 -> p_s, 1024..2047 -> p_t.
// ---------------------------------------------------------------------------
__global__ void softmax_kernel(const float* __restrict__ y_s,
                               const float* __restrict__ y_t,
                               float* __restrict__ p_s,
                               float* __restrict__ p_t) {
    const int   row = blockIdx.x & 1023;
    const float* y  = (blockIdx.x < 1024) ? y_s : y_t;
    float*       p  = (blockIdx.x < 1024) ? p_s : p_t;
    const float* yr = y + (size_t)row * Vn;
    float*       pr = p + (size_t)row * Vn;
    const int tid = threadIdx.x;
    __shared__ float red[256];

    float m = -3.4e38f;
    for (int i = tid; i < Vn; i += 256) m = fmaxf(m, yr[i] * 0.5f);
    red[tid] = m; __syncthreads();
    for (int s = 128; s > 0; s >>= 1) {
        if (tid < s) red[tid] = fmaxf(red[tid], red[tid + s]);
        __syncthreads();
    }
    m = red[0]; __syncthreads();

    float sum = 0.f;
    for (int i = tid; i < Vn; i += 256) {
        float e = __expf(yr[i] * 0.5f - m);
        pr[i] = e; sum += e;
    }
    red[tid] = sum; __syncthreads();
    for (int s = 128; s > 0; s >>= 1) {
        if (tid < s) red[tid] += red[tid + s];
        __syncthreads();
    }
    const float inv = 1.f / red[0];
    for (int i = tid; i < Vn; i += 256) pr[i] *= inv;
}

// ---------------------------------------------------------------------------
// 2) L1 cdist (the FLOP-dominant part): W[m][n] = sum_v |p_s[m,v]-p_t[n,v]|
//    and P = K = exp(-W/0.1).  64x64 tile / block, 4x4 per thread, BK=16.
//    fp32 VALU bound; all tile loads are L2 hits (32MB working set << 192MB).
// ---------------------------------------------------------------------------
#define BM 64
#define BK 16
#define LDT 72   // padded LDS stride -> conflict-free & 16B aligned (72*4=288)

__global__ void cdist_kernel(const float* __restrict__ p_s,
                             const float* __restrict__ p_t,
                             float* __restrict__ W,
                             float* __restrict__ P) {
    __shared__ float As[BK][LDT];
    __shared__ float Bs[BK][LDT];
    const int tid = threadIdx.x;          // 256 threads
    const int tx  = tid & 15;             // 16 x 16 thread tile
    const int ty  = tid >> 4;
    const int m0  = blockIdx.x * BM;
    const int n0  = blockIdx.y * BM;
    const int lk  = tid & 15;             // loader: k within tile
    const int lm  = tid >> 4;             // loader: base row within tile

    float acc[4][4] = {};

    for (int k0 = 0; k0 < Vn; k0 += BK) {
#pragma unroll
        for (int r = 0; r < 4; ++r) {
            const int m = lm + r * 16;
            As[lk][m] = p_s[(size_t)(m0 + m) * Vn + k0 + lk];
            Bs[lk][m] = p_t[(size_t)(n0 + m) * Vn + k0 + lk];
        }
        if (k0 + BK < Vn) {   // hint next tile into cache (global_prefetch_b8)
            __builtin_prefetch(&p_s[(size_t)(m0 + lm) * Vn + k0 + BK + lk], 0, 0);
            __builtin_prefetch(&p_t[(size_t)(n0 + lm) * Vn + k0 + BK + lk], 0, 0);
        }
        __syncthreads();
#pragma unroll
        for (int k = 0; k < BK; ++k) {
            float a[4], b[4];
#pragma unroll
            for (int i = 0; i < 4; ++i) a[i] = As[k][ty * 4 + i];
#pragma unroll
            for (int j = 0; j < 4; ++j) b[j] = Bs[k][tx * 4 + j];
#pragma unroll
            for (int i = 0; i < 4; ++i)
#pragma unroll
                for (int j = 0; j < 4; ++j)
                    acc[i][j] += fabsf(a[i] - b[j]);   // v_sub + v_add |mod|
        }
        __syncthreads();
    }

#pragma unroll
    for (int i = 0; i < 4; ++i)
#pragma unroll
        for (int j = 0; j < 4; ++j) {
            const int m = m0 + ty * 4 + i;
            const int n = n0 + tx * 4 + j;
            const float w = acc[i][j];
            W[(size_t)m * Bn + n] = w;
            P[(size_t)m * Bn + n] = __expf(-10.f * w);   // K = exp(-W/eps)
        }
}

// ---------------------------------------------------------------------------
// 3a) Row normalize.  Row sums computed as P x ones via V_WMMA_F32_16X16X4_F32
//     (wave 0 of each block; 16 rows per block, K accumulated 4 at a time).
//     A layout per ISA 16x4 fp32 table: lane<16 -> K={0,1}, lane>=16 -> K={2,3}.
//     B = all-ones so its layout is irrelevant; every column of D = rowsum.
// ---------------------------------------------------------------------------
__global__ void rownorm_kernel(float* __restrict__ P) {
    __shared__ float rsum[16];
    const int tid     = threadIdx.x;       // 256
    const int rowbase = blockIdx.x * 16;   // 64 blocks

    if (tid < 32) {                         // exactly wave 0, EXEC all ones
        const int lane = tid;
        const int half = lane >> 4;         // 0: K=0,1   1: K=2,3
        const int row  = rowbase + (lane & 15);
        const float* pr = P + (size_t)row * Bn + 2 * half;
        v2f b; b.x = 1.f; b.y = 1.f;        // ones matrix
        v8f c = {};
        for (int k = 0; k < Bn; k += 4) {
            v2f a;
            a.x = pr[k];
            a.y = pr[k + 1];
            c = __builtin_amdgcn_wmma_f32_16x16x4_f32(
                    false, a, false, b, (short)0, c, false, false);
        }
        // C/D layout: vgpr r -> M=r (lanes 0-15) or M=8+r (lanes 16-31)
        if ((lane & 15) == 0) {
#pragma unroll
            for (int r = 0; r < 8; ++r) rsum[8 * half + r] = c[r];
        }
    }
    __syncthreads();

    for (int idx = tid; idx < 16 * Bn; idx += 256) {
        const int r = idx >> 10;
        P[(size_t)(rowbase + r) * Bn + (idx & 1023)] /= rsum[r];
    }
}

// ---------------------------------------------------------------------------
// 3b) Column sums (thread-per-column is perfectly coalesced), then divide.
// ---------------------------------------------------------------------------
__global__ void colsum_kernel(const float* __restrict__ P,
                              float* __restrict__ cs) {
    const int j = blockIdx.x * 256 + threadIdx.x;   // 4 blocks -> 1024 cols
    float s = 0.f;
#pragma unroll 8
    for (int i = 0; i < Bn; ++i) s += P[(size_t)i * Bn + j];
    cs[j] = s;
}

__global__ void coldiv_kernel(float* __restrict__ P,
                              const float* __restrict__ cs) {
    const int idx = blockIdx.x * 256 + threadIdx.x; // 4096 blocks -> 1M elems
    P[idx] = P[idx] / cs[idx & 1023];
}

// ---------------------------------------------------------------------------
// 4) out = 0.001 * sum(P * W)   (deterministic two-stage tree reduction)
// ---------------------------------------------------------------------------
__global__ void dotred_kernel(const float* __restrict__ P,
                              const float* __restrict__ W,
                              float* __restrict__ partial) {
    __shared__ float red[256];
    const int tid = threadIdx.x;
    const size_t base = (size_t)blockIdx.x * 4096;   // 256 blocks
    float s = 0.f;
    for (int i = tid; i < 4096; i += 256) s += P[base + i] * W[base + i];
    red[tid] = s; __syncthreads();
    for (int st = 128; st > 0; st >>= 1) {
        if (tid < st) red[tid] += red[tid + st];
        __syncthreads();
    }
    if (tid == 0) partial[blockIdx.x] = red[0];
}

__global__ void final_kernel(const float* __restrict__ partial,
                             float* __restrict__ out) {
    __shared__ float red[256];
    const int tid = threadIdx.x;
    red[tid] = partial[tid]; __syncthreads();
    for (int st = 128; st > 0; st >>= 1) {
        if (tid < st) red[tid] += red[tid + st];
        __syncthreads();
    }
    if (tid == 0) out[0] = 0.001f * red[0];
}

// ---------------------------------------------------------------------------
extern "C" void kernel_launch(void* const* d_in, const int* in_sizes, int n_in,
                              void* d_out, int out_size, void* d_ws, size_t ws_size,
                              hipStream_t stream) {
    const float* y_s = (const float*)d_in[0];
    const float* y_t = (const float*)d_in[1];

    float* ws      = (float*)d_ws;
    float* p_s     = ws;                              // 1024*4096
    float* p_t     = p_s + (size_t)Bn * Vn;           // 1024*4096
    float* W       = p_t + (size_t)Bn * Vn;           // 1024*1024
    float* P       = W   + (size_t)Bn * Bn;           // 1024*1024
    float* cs      = P   + (size_t)Bn * Bn;           // 1024
    float* partial = cs  + Bn;                        // 256

    softmax_kernel<<<2048, 256, 0, stream>>>(y_s, y_t, p_s, p_t);
    cdist_kernel<<<dim3(16, 16), 256, 0, stream>>>(p_s, p_t, W, P);

    for (int it = 0; it < 20; ++it) {
        rownorm_kernel<<<64, 256, 0, stream>>>(P);
        colsum_kernel<<<4, 256, 0, stream>>>(P, cs);
        coldiv_kernel<<<4096, 256, 0, stream>>>(P, cs);
    }

    dotred_kernel<<<256, 256, 0, stream>>>(P, W, partial);
    final_kernel<<<1, 256, 0, stream>>>(partial, (float*)d_out);
}